// Adaptive_Wavelet_Tranform_85375359910610
// MI455X (gfx1250) — compile-verified
//
#include <hip/hip_runtime.h>

typedef __attribute__((ext_vector_type(2))) float f32x2;
typedef __attribute__((ext_vector_type(8))) float f32x8;

#define T_LEN 4096
#define C_CH  512
#define B_N   32
#define W_OUT 2055
#define WP    2064          // padded row stride for low/high (64B aligned, 16B-safe overread)
#define R_ROWS (B_N * C_CH) // 16384

__constant__ float c_dlo[16] = {
  -0.00011747678400228192f,  0.0006754494059985568f,
  -0.0003917403729959771f,  -0.00487035299301066f,
   0.008746094047015655f,    0.013981027917015516f,
  -0.04408825393106472f,    -0.01736930100202211f,
   0.128747426620186f,       0.00047248457399797254f,
  -0.2840155429624281f,     -0.015829105256023893f,
   0.5853546836548691f,      0.6756307362980128f,
   0.3128715909144659f,      0.05441584224308161f
};

// ---- CDNA5 async LDS data path -------------------------------------------
__device__ __forceinline__ unsigned lds_off(const void* p) {
  // low 32 bits of a flat shared-memory pointer == LDS byte offset
  return (unsigned)(size_t)p;
}
__device__ __forceinline__ void async_ld_b32(unsigned lds, const float* g) {
  asm volatile("global_load_async_to_lds_b32 %0, %1, off"
               :: "v"(lds), "v"(g) : "memory");
}
__device__ __forceinline__ void async_ld_b128(unsigned lds, const float* g) {
  asm volatile("global_load_async_to_lds_b128 %0, %1, off"
               :: "v"(lds), "v"(g) : "memory");
}
__device__ __forceinline__ void async_st_b32(float* g, unsigned lds) {
  asm volatile("global_store_async_from_lds_b32 %0, %1, off"
               :: "v"(g), "v"(lds) : "memory");
}
__device__ __forceinline__ void wait_async0() {
#if __has_builtin(__builtin_amdgcn_s_wait_asynccnt)
  __builtin_amdgcn_s_wait_asynccnt(0);
#else
  asm volatile("s_wait_asynccnt 0x0" ::: "memory");
#endif
}

// symmetric extension index (pywt 'symmetric'): valid for q in [-T, 2T-1]
__device__ __forceinline__ int symidx(int q) {
  if (q < 0) q = -q - 1;
  if (q >= T_LEN) q = 2 * T_LEN - 1 - q;
  return q;
}

// DWT analysis filter matrix A[m][kg]: row0 = dec_lo reversed, row1 = dec_hi reversed
__device__ __forceinline__ float dwtA(int m, int kg) {
  if (m == 0) return c_dlo[15 - kg];
  if (m == 1) return (kg & 1) ? -c_dlo[kg] : c_dlo[kg]; // dec_hi[15-kg] = (-1)^kg * dec_lo[kg]
  return 0.f;
}

// IDWT synthesis filter matrix A[m][kg]: m=0 even-t phase, m=1 odd-t phase;
// kg<8 -> ca taps, kg>=8 -> cd taps
__device__ __forceinline__ float idwtA(int m, int kg) {
  if (m == 0) return (kg < 8) ? c_dlo[2 * kg + 1] :  c_dlo[30 - 2 * kg];
  if (m == 1) return (kg < 8) ? c_dlo[2 * kg]     : -c_dlo[31 - 2 * kg];
  return 0.f;
}

// ---------------------------------------------------------------------------
// Kernel 1: transpose-read (async to LDS) + db8 DWT via V_WMMA_F32_16X16X4_F32
// block: 256 threads; tile = (b, 32 channels, 64 output positions)
// ---------------------------------------------------------------------------
__global__ __launch_bounds__(256) void dwt_kernel(const float* __restrict__ x,
                                                  float* __restrict__ low,
                                                  float* __restrict__ high) {
  __shared__ float eData[32][144]; // 142 window samples per channel (+pad)

  int bid = blockIdx.x;
  int jt = bid % 33;
  int ct = (bid / 33) & 15;
  int b  = bid / (33 * 16);
  int j0 = jt * 64;
  int c0 = ct * 32;

  int tid = threadIdx.x;
  int c_l = tid & 31;
  const float* xb = x + (size_t)b * T_LEN * C_CH + (size_t)(c0 + c_l);
  for (int t_l = (tid >> 5); t_l < 142; t_l += 8) {
    int ts = symidx(2 * j0 - 14 + t_l);
    // per-lane async copy: 32 lanes = one 128B contiguous global read, LDS dest
    // implements the (t,c) transpose for free
    async_ld_b32(lds_off(&eData[c_l][t_l]), xb + (size_t)ts * C_CH);
  }
  wait_async0();
  __syncthreads();

  int wave = tid >> 5;
  int lane = tid & 31;
  int n  = lane & 15;
  int hi = lane >> 4;

  // A operand per K-chunk (depends only on lane)
  f32x2 a4[4];
#pragma unroll
  for (int c4 = 0; c4 < 4; ++c4) {
    int kb = c4 * 4 + hi * 2;
    a4[c4].x = dwtA(n, kb);
    a4[c4].y = dwtA(n, kb + 1);
  }

  // 128 tasks = 32 channels x 4 sub-tiles of 16 outputs; 16 per wave
  for (int it = 0; it < 16; ++it) {
    int task = wave * 16 + it;
    int ch = task & 31;
    int js = task >> 5; // 0..3
    f32x8 acc = {};
#pragma unroll
    for (int c4 = 0; c4 < 4; ++c4) {
      int kb = c4 * 4 + hi * 2;
      int eb = 2 * (js * 16 + n) + kb;
      f32x2 bm;
      bm.x = eData[ch][eb];
      bm.y = eData[ch][eb + 1];
      acc = __builtin_amdgcn_wmma_f32_16x16x4_f32(false, a4[c4], false, bm,
                                                  (short)0, acc, false, false);
    }
    if (hi == 0) {
      int j = j0 + js * 16 + n;
      if (j < W_OUT) {
        size_t row = (size_t)(b * C_CH + c0 + ch) * WP;
        low[row + j]  = acc[0]; // D row 0 (lanes 0-15, VGPR0)
        high[row + j] = acc[1]; // D row 1 (lanes 0-15, VGPR1)
      }
    }
  }
}

// ---------------------------------------------------------------------------
// Kernel 2: per-row quantile of high^2 via in-LDS bitonic sort (pad +INF)
// one 256-thread block per row; row staged with async b128
// ---------------------------------------------------------------------------
__global__ __launch_bounds__(256) void thr_kernel(const float* __restrict__ high,
                                                  const float* __restrict__ qp,
                                                  float* __restrict__ thr) {
  __shared__ float s[4096];
  int r = blockIdx.x;
  const float* hrow = high + (size_t)r * WP; // 16B aligned (WP*4 = 8256)
  for (int idx = threadIdx.x; idx < 514; idx += 256) { // 514*4 = 2056 >= W_OUT
    async_ld_b128(lds_off(&s[idx * 4]), hrow + idx * 4);
  }
  wait_async0();
  __syncthreads();
  for (int i = threadIdx.x; i < 4096; i += 256) {
    if (i < W_OUT) {
      float v = s[i];
      s[i] = v * v;
    } else {
      s[i] = __builtin_inff();
    }
  }
  __syncthreads();

  for (int k = 2; k <= 4096; k <<= 1) {
    for (int j = k >> 1; j > 0; j >>= 1) {
      for (int i = threadIdx.x; i < 4096; i += 256) {
        int ixj = i ^ j;
        if (ixj > i) {
          float a = s[i], b2 = s[ixj];
          bool up = ((i & k) == 0);
          if ((a > b2) == up) { s[i] = b2; s[ixj] = a; }
        }
      }
      __syncthreads();
    }
  }

  if (threadIdx.x == 0) {
    float q = fminf(fmaxf(qp[0], 0.f), 1.f);
    float h = q * (float)(W_OUT - 1);
    int k0 = (int)floorf(h);
    if (k0 > W_OUT - 1) k0 = W_OUT - 1;
    int k1 = (k0 + 1 < W_OUT) ? k0 + 1 : k0;
    float g = h - (float)k0;
    thr[r] = s[k0] + g * (s[k1] - s[k0]);
  }
}

// ---------------------------------------------------------------------------
// Kernel 3: mask + db8 IDWT via WMMA; async-staged coeffs, async store of output
// block: 256 threads; tile = (b, 32 channels, 64 time samples)
// ---------------------------------------------------------------------------
__global__ __launch_bounds__(256) void idwt_kernel(const float* __restrict__ low,
                                                   const float* __restrict__ high,
                                                   const float* __restrict__ thr,
                                                   float* __restrict__ out) {
  __shared__ float caS[32][40];
  __shared__ float cdS[32][40];
  __shared__ float thS[32];
  __shared__ float recS[32][64];

  int bid = blockIdx.x;
  int tt = bid & 63;
  int ct = (bid >> 6) & 15;
  int b  = bid >> 10;
  int t0 = tt * 64;
  int c0 = ct * 32;
  int m0 = t0 >> 1; // multiple of 32 -> 16B aligned global offsets

  int tid = threadIdx.x;
  if (tid < 32) thS[tid] = thr[b * C_CH + c0 + tid];

  // stage 32 rows x 40 coeffs of ca and cd: 10 b128 chunks per row per array.
  // m0+39 can reach 2055 (> W_OUT-1) only in the last tile; the overread lands
  // in the WP-row padding and those slots are never referenced by compute.
  for (int idx = tid; idx < 32 * 10; idx += 256) {
    int ch = idx / 10;
    int q4 = (idx - ch * 10) * 4;
    size_t row = (size_t)(b * C_CH + c0 + ch) * WP + (size_t)m0 + (size_t)q4;
    async_ld_b128(lds_off(&caS[ch][q4]), low + row);
    async_ld_b128(lds_off(&cdS[ch][q4]), high + row);
  }
  wait_async0();
  __syncthreads();

  // hard threshold high coeffs in LDS: keep iff cd^2 > thr (strict)
  for (int idx = tid; idx < 32 * 40; idx += 256) {
    int ch = idx / 40;
    int ml = idx - ch * 40;
    float v = cdS[ch][ml];
    cdS[ch][ml] = (v * v > thS[ch]) ? v : 0.f;
  }
  __syncthreads();

  int wave = tid >> 5;
  int lane = tid & 31;
  int n  = lane & 15;
  int hi = lane >> 4;

  f32x2 a4[4];
#pragma unroll
  for (int c4 = 0; c4 < 4; ++c4) {
    int kb = c4 * 4 + hi * 2;
    a4[c4].x = idwtA(n, kb);
    a4[c4].y = idwtA(n, kb + 1);
  }

  // 64 tasks = 32 channels x 2 chains of 32 samples; 8 per wave
  for (int it = 0; it < 8; ++it) {
    int task = wave * 8 + it;
    int ch   = task & 31;
    int half = task >> 5; // 0..1
    int mb   = half * 16;
    f32x8 acc = {};
#pragma unroll
    for (int c4 = 0; c4 < 4; ++c4) {
      int kb = c4 * 4 + hi * 2; // even; kb and kb+1 stay within ca(<8) or cd(>=8)
      f32x2 bm;
      if (kb < 8) {
        bm.x = caS[ch][mb + n + kb];
        bm.y = caS[ch][mb + n + kb + 1];
      } else {
        bm.x = cdS[ch][mb + n + (kb - 8)];
        bm.y = cdS[ch][mb + n + (kb - 7)];
      }
      acc = __builtin_amdgcn_wmma_f32_16x16x4_f32(false, a4[c4], false, bm,
                                                  (short)0, acc, false, false);
    }
    if (hi == 0) {
      recS[ch][half * 32 + 2 * n]     = acc[0]; // even t
      recS[ch][half * 32 + 2 * n + 1] = acc[1]; // odd t
    }
  }
  __syncthreads();

  // transposed output: LDS -> global without VGPR round-trip; implicit
  // wait-idle at S_ENDPGM covers ASYNCcnt completion.
  int c_l = tid & 31;
  for (int t_l = (tid >> 5); t_l < 64; t_l += 8) {
    float* g = out + ((size_t)b * T_LEN + (size_t)(t0 + t_l)) * C_CH +
               (size_t)(c0 + c_l);
    async_st_b32(g, lds_off(&recS[c_l][t_l]));
  }
}

// ---------------------------------------------------------------------------
extern "C" void kernel_launch(void* const* d_in, const int* in_sizes, int n_in,
                              void* d_out, int out_size, void* d_ws, size_t ws_size,
                              hipStream_t stream) {
  const float* x  = (const float*)d_in[0];
  const float* qp = (const float*)d_in[1];
  float* out = (float*)d_out;

  float* low  = (float*)d_ws;
  float* high = low + (size_t)R_ROWS * WP;
  float* thr  = high + (size_t)R_ROWS * WP;

  dwt_kernel<<<dim3(B_N * 16 * 33), 256, 0, stream>>>(x, low, high);
  thr_kernel<<<dim3(R_ROWS), 256, 0, stream>>>(high, qp, thr);
  idwt_kernel<<<dim3(B_N * 16 * 64), 256, 0, stream>>>(low, high, thr, out);
}